// StepSeqFastWeightsAttention_22866405884269
// MI455X (gfx1250) — compile-verified
//
#include <hip/hip_runtime.h>
#include <hip/hip_bf16.h>
#include <math.h>

// ---------------------------------------------------------------------------
// StepSeqFastWeightsAttention for MI455X (gfx1250, wave32)
//
// Roofline: weights state is 512 MB read + 512 MB write => ~44 us @ 23.3 TB/s.
// Everything else (4 GEMMs of 256x1024x1024, dpfp, sigmoid) is a few us and
// fits in the 192 MB L2. So:
//   * k_fw_update: single streaming pass over weights; computes W.phik and
//     W.phiq per row, delta = beta*(v - W.phik), writes W + delta x phik from
//     registers, and emits out_head = W.phiq + delta*(phik.phiq) (algebraic
//     fusion -> no second pass over the 512 MB state).
//   * GEMMs use V_WMMA_F32_16X16X4_F32 (exact f32, CDNA5 matrix pipe).
// ---------------------------------------------------------------------------

typedef float v2f __attribute__((ext_vector_type(2)));
typedef float v8f __attribute__((ext_vector_type(8)));

#define NHEADS 16
#define DMODEL 1024
#define DK     64
#define DPHI   512

// ---------------------------------------------------------------------------
// Kernel 1: projections  qp/kp/vp = x @ {Wq,Wk,Wv}   (f32 WMMA 16x16x4)
// One wave computes one 16x16 C tile, K-loop in steps of 4.
// A frag: lane(0..15)=M, lane half selects K pair -> contiguous float2 load.
// B frag (assumed mirror of A): lane=N, lane half selects K pair.
// ---------------------------------------------------------------------------
__global__ void k_proj_wmma(const float* __restrict__ X,
                            const float* __restrict__ Wq,
                            const float* __restrict__ Wk,
                            const float* __restrict__ Wv,
                            float* __restrict__ qp,
                            float* __restrict__ kp,
                            float* __restrict__ vp)
{
    const float* W;
    float* O;
    if (blockIdx.z == 0)      { W = Wq; O = qp; }
    else if (blockIdx.z == 1) { W = Wk; O = kp; }
    else                      { W = Wv; O = vp; }

    const int lane = threadIdx.x;            // 0..31
    const int lm = lane & 15, lh = lane >> 4;
    const int n0 = blockIdx.x * 16;
    const int m0 = (blockIdx.y * blockDim.y + threadIdx.y) * 16;

    const float* ap = X + (size_t)(m0 + lm) * DMODEL + 2 * lh;
    const float* bp = W + (size_t)(2 * lh) * DMODEL + n0 + lm;

    v8f c = {};
    for (int k0 = 0; k0 < DMODEL; k0 += 4) {
        v2f a = *(const v2f*)(ap + k0);            // A[m][k0+2lh .. +1]
        v2f b;
        b.x = bp[(size_t)k0 * DMODEL];             // B[k0+2lh][n]
        b.y = bp[(size_t)(k0 + 1) * DMODEL];       // B[k0+2lh+1][n]
        c = __builtin_amdgcn_wmma_f32_16x16x4_f32(false, a, false, b,
                                                  (short)0, c, false, false);
    }
    const int row_base = m0 + lh * 8;
    const int col = n0 + lm;
#pragma unroll
    for (int i = 0; i < 8; ++i)
        O[(size_t)(row_base + i) * DMODEL + col] = c[i];
}

// ---------------------------------------------------------------------------
// Kernel 2: beta = sigmoid(x @ Wg), one wave per (b,h)
// ---------------------------------------------------------------------------
__global__ void k_beta(const float* __restrict__ X,
                       const float* __restrict__ Wg,
                       float* __restrict__ beta, int BH)
{
    const int w = (blockIdx.x * blockDim.x + threadIdx.x) >> 5;
    const int lane = threadIdx.x & 31;
    if (w >= BH) return;
    const int b = w >> 4, h = w & 15;
    const float* xp = X + (size_t)b * DMODEL;
    float s = 0.f;
    for (int j = lane; j < DMODEL; j += 32)
        s += xp[j] * Wg[j * NHEADS + h];
#pragma unroll
    for (int m = 16; m >= 1; m >>= 1) s += __shfl_xor(s, m, 32);
    if (lane == 0) beta[w] = 1.f / (1.f + expf(-s));
}

// ---------------------------------------------------------------------------
// Kernel 3: dpfp.  One 128-thread block per (b,h).
// x2 = relu([d,-d]) (128);  phi[p] = x2[p%128]*x2[(p%128 - (p/128+1)) & 127]
// normalized by (sum + 1e-6).
// ---------------------------------------------------------------------------
__global__ void k_dpfp(const float* __restrict__ proj,
                       float* __restrict__ phi)
{
    const int bh = blockIdx.x;
    const int t = threadIdx.x;                 // 0..127
    __shared__ float x2[128];
    __shared__ float red[128];

    const float* d = proj + (size_t)bh * 2 * DK;  // 128 values per (b,h)? no: 64
    // proj layout: [b, h*64 .. h*64+64) within row of 1024 -> flat bh*64
    const float* dd = proj + (size_t)bh * DK;
    float val = (t < DK) ? dd[t] : -dd[t - DK];
    (void)d;
    x2[t] = fmaxf(val, 0.f);
    __syncthreads();

    const float xt = x2[t];
    float p[4];
    float s = 0.f;
#pragma unroll
    for (int r = 1; r <= 4; ++r) {
        float v = xt * x2[(t - r) & 127];
        p[r - 1] = v;
        s += v;
    }
    red[t] = s;
    __syncthreads();
    for (int off = 64; off >= 1; off >>= 1) {
        if (t < off) red[t] += red[t + off];
        __syncthreads();
    }
    const float inv = 1.f / (red[0] + 1e-6f);
    float* o = phi + (size_t)bh * DPHI;
#pragma unroll
    for (int r = 0; r < 4; ++r) o[t + 128 * r] = p[r] * inv;
}

// ---------------------------------------------------------------------------
// Kernel 4: the bandwidth-bound fast-weight pass. One 256-thread block (8
// wave32) per (b,h). Each wave owns 8 rows of the 64x512 state. Single read
// + single write of the 512 MB state; attn head output fused in.
// ---------------------------------------------------------------------------
__global__ void k_fw_update(const float* __restrict__ Wold,
                            const float* __restrict__ phiK,
                            const float* __restrict__ phiQ,
                            const float* __restrict__ Vp,
                            const float* __restrict__ Beta,
                            float* __restrict__ Wnew,
                            float* __restrict__ attn)
{
    const int bh = blockIdx.x;
    const int t = threadIdx.x;                 // 0..255
    const int lane = t & 31, wid = t >> 5;     // 8 waves

    __shared__ __align__(16) float pk[DPHI];
    __shared__ __align__(16) float pq[DPHI];
    __shared__ float vh[DK];
    __shared__ float red[256];
    __shared__ float s_dot, s_beta;

    const size_t pbase = (size_t)bh * DPHI;
    pk[t]       = phiK[pbase + t];
    pk[t + 256] = phiK[pbase + 256 + t];
    pq[t]       = phiQ[pbase + t];
    pq[t + 256] = phiQ[pbase + 256 + t];
    if (t < DK) vh[t] = Vp[(size_t)bh * DK + t];
    __syncthreads();

    // s = phik . phiq (needed for fused read-with-query)
    red[t] = pk[t] * pq[t] + pk[t + 256] * pq[t + 256];
    __syncthreads();
    for (int off = 128; off >= 1; off >>= 1) {
        if (t < off) red[t] += red[t + off];
        __syncthreads();
    }
    if (t == 0) { s_dot = red[0]; s_beta = Beta[bh]; }
    __syncthreads();
    const float s = s_dot, beta = s_beta;

    const float* win  = Wold + (size_t)bh * DK * DPHI;
    float*       wout = Wnew + (size_t)bh * DK * DPHI;
    const float4* pk4 = (const float4*)pk;
    const float4* pq4 = (const float4*)pq;

    for (int rr = 0; rr < 8; ++rr) {
        const int row = wid * 8 + rr;
        const float4* ip = (const float4*)(win + (size_t)row * DPHI);
        float4*       op = (float4*)(wout + (size_t)row * DPHI);

        float4 w[4];
        float dk = 0.f, dq = 0.f;
#pragma unroll
        for (int c = 0; c < 4; ++c) {
            const int j = lane + 32 * c;       // coalesced float4 per lane
            w[c] = ip[j];
            const float4 k4 = pk4[j];
            const float4 q4 = pq4[j];
            dk += w[c].x * k4.x + w[c].y * k4.y + w[c].z * k4.z + w[c].w * k4.w;
            dq += w[c].x * q4.x + w[c].y * q4.y + w[c].z * q4.z + w[c].w * q4.w;
        }
#pragma unroll
        for (int m = 16; m >= 1; m >>= 1) {
            dk += __shfl_xor(dk, m, 32);
            dq += __shfl_xor(dq, m, 32);
        }
        const float delta = beta * (vh[row] - dk);
#pragma unroll
        for (int c = 0; c < 4; ++c) {
            const int j = lane + 32 * c;
            const float4 k4 = pk4[j];
            float4 nw;
            nw.x = w[c].x + delta * k4.x;
            nw.y = w[c].y + delta * k4.y;
            nw.z = w[c].z + delta * k4.z;
            nw.w = w[c].w + delta * k4.w;
            op[j] = nw;
        }
        if (lane == 0) {
            const int b = bh >> 4, h = bh & 15;
            // out_head = newW.phiq = W.phiq + delta*(phik.phiq)
            attn[(size_t)b * DMODEL + h * DK + row] = dq + delta * s;
        }
    }
}

// ---------------------------------------------------------------------------
// Kernel 5: out = attn @ Wo^T + bo  (f32 WMMA; B-frag K pairs are contiguous
// in Wo's rows since B'[k][n] = Wo[n][k]).
// ---------------------------------------------------------------------------
__global__ void k_out_wmma(const float* __restrict__ attn,
                           const float* __restrict__ Wo,
                           const float* __restrict__ bo,
                           float* __restrict__ out)
{
    const int lane = threadIdx.x;
    const int lm = lane & 15, lh = lane >> 4;
    const int n0 = blockIdx.x * 16;
    const int m0 = (blockIdx.y * blockDim.y + threadIdx.y) * 16;

    const float* ap = attn + (size_t)(m0 + lm) * DMODEL + 2 * lh;
    const float* bp = Wo   + (size_t)(n0 + lm) * DMODEL + 2 * lh;

    v8f c = {};
    for (int k0 = 0; k0 < DMODEL; k0 += 4) {
        v2f a = *(const v2f*)(ap + k0);
        v2f b = *(const v2f*)(bp + k0);   // Wo[n][k0+2lh], Wo[n][k0+2lh+1]
        c = __builtin_amdgcn_wmma_f32_16x16x4_f32(false, a, false, b,
                                                  (short)0, c, false, false);
    }
    const float bias = bo[n0 + lm];
    const int row_base = m0 + lh * 8;
    const int col = n0 + lm;
#pragma unroll
    for (int i = 0; i < 8; ++i)
        out[(size_t)(row_base + i) * DMODEL + col] = c[i] + bias;
}

// ---------------------------------------------------------------------------
extern "C" void kernel_launch(void* const* d_in, const int* in_sizes, int n_in,
                              void* d_out, int out_size, void* d_ws, size_t ws_size,
                              hipStream_t stream)
{
    const float* x       = (const float*)d_in[0];
    const float* weights = (const float*)d_in[1];
    const float* Wq      = (const float*)d_in[2];
    const float* Wk      = (const float*)d_in[3];
    const float* Wv      = (const float*)d_in[4];
    const float* Wg      = (const float*)d_in[5];
    const float* Wo      = (const float*)d_in[6];
    const float* bo      = (const float*)d_in[7];

    const int B  = in_sizes[0] / DMODEL;     // 256
    const int BH = B * NHEADS;               // 4096

    float* out   = (float*)d_out;                       // [B,1024]
    float* new_w = out + (size_t)B * DMODEL;            // [B,16,64,512]

    // workspace layout (floats): ~21 MB total
    float* qp    = (float*)d_ws;                        // [B,1024]
    float* kp    = qp   + (size_t)B * DMODEL;
    float* vp    = kp   + (size_t)B * DMODEL;
    float* beta  = vp   + (size_t)B * DMODEL;           // [B*16]
    float* phiq  = beta + (size_t)BH;                   // [B*16,512]
    float* phik  = phiq + (size_t)BH * DPHI;
    float* attn  = phik + (size_t)BH * DPHI;            // [B,1024]

    // 1) projections (WMMA f32)
    k_proj_wmma<<<dim3(DMODEL / 16, B / 64, 3), dim3(32, 4), 0, stream>>>(
        x, Wq, Wk, Wv, qp, kp, vp);

    // 2) gate
    k_beta<<<(BH * 32 + 127) / 128, 128, 0, stream>>>(x, Wg, beta, BH);

    // 3) dpfp feature maps
    k_dpfp<<<BH, 128, 0, stream>>>(qp, phiq);
    k_dpfp<<<BH, 128, 0, stream>>>(kp, phik);

    // 4) bandwidth-bound fast-weight read/update/read pass (the 1 GB pass)
    k_fw_update<<<BH, 256, 0, stream>>>(weights, phik, phiq, vp, beta,
                                        new_w, attn);

    // 5) output projection + bias (WMMA f32)
    k_out_wmma<<<dim3(DMODEL / 16, B / 64), dim3(32, 4), 0, stream>>>(
        attn, Wo, bo, out);
}